// CustomGATLayer_13984413516293
// MI455X (gfx1250) — compile-verified
//
#include <hip/hip_runtime.h>
#include <math.h>

#define IN_DIM   128
#define HEADS    4
#define OUT_FEATS 32
#define NEG_SLOPE 0.2f

typedef __attribute__((ext_vector_type(2))) float v2f;
typedef __attribute__((ext_vector_type(8))) float v8f;

// ---------------------------------------------------------------------------
// K1: hf = h @ fc_w^T   via V_WMMA_F32_16X16X4_F32 (exact f32, K=4 per step)
// Block = 128 threads (4 waves). Each block computes a 16-row tile of hf;
// wave w computes column tiles 2w and 2w+1 (8 tiles cover 128 cols).
// A tile (16x128) staged in LDS with stride 132 (bank-conflict-free frags).
// B fragments (fc_w rows) read from global: fc_w is 64KB -> L2 resident.
// ---------------------------------------------------------------------------
__global__ void __launch_bounds__(128) gat_gemm_wmma(
    const float* __restrict__ h, const float* __restrict__ fc_w,
    float* __restrict__ hf)
{
    __shared__ float lds_a[16 * 132];
    const int tile_row = blockIdx.x;      // 16 rows per block
    const int t = threadIdx.x;            // 0..127

    // stage A tile: thread t loads 16 floats of row t/8, col base (t%8)*16
    {
        const int row  = t >> 3;
        const int colb = (t & 7) * 16;
        const float4* src4 =
            (const float4*)(h + (size_t)(tile_row * 16 + row) * IN_DIM + colb);
        float4 r0 = src4[0], r1 = src4[1], r2 = src4[2], r3 = src4[3];
        float4* dst4 = (float4*)(lds_a + row * 132 + colb);
        dst4[0] = r0; dst4[1] = r1; dst4[2] = r2; dst4[3] = r3;
    }
    __syncthreads();

    const int wave = t >> 5;
    const int lane = t & 31;
    const int half = lane >> 4;   // 0: K pair {0,1}, 1: K pair {2,3}
    const int l16  = lane & 15;

    for (int ct = 0; ct < 2; ++ct) {
        const int j0 = (wave * 2 + ct) * 16;   // output column base
        v8f c = {};
#pragma unroll
        for (int kk = 0; kk < 32; ++kk) {      // 32 x K=4  => K=128
            const int kbase = kk * 4 + half * 2;
            v2f a, b;
            // A 16x4 fragment: lane l16 holds row l16, K = kbase, kbase+1
            const float* ap = lds_a + l16 * 132 + kbase;
            a.x = ap[0]; a.y = ap[1];
            // B 4x16 fragment: B[k][j] = fc_w[j][k]
            const float* bp = fc_w + (size_t)(j0 + l16) * IN_DIM + kbase;
            b.x = bp[0]; b.y = bp[1];
            c = __builtin_amdgcn_wmma_f32_16x16x4_f32(
                    false, a, false, b, (short)0, c, false, false);
        }
        // C/D layout: VGPR r -> rows r (lanes 0-15) and r+8 (lanes 16-31)
        float* outp = hf + ((size_t)tile_row * 16 + half * 8) * IN_DIM + j0 + l16;
#pragma unroll
        for (int r = 0; r < 8; ++r)
            outp[(size_t)r * IN_DIM] = c[r];
    }
}

// ---------------------------------------------------------------------------
// K2: el/er per-node attention halves. Block = 128 = 4 waves; wave w = head w.
// ---------------------------------------------------------------------------
__global__ void __launch_bounds__(128) gat_elr(
    const float* __restrict__ hf, const float* __restrict__ attn_l,
    const float* __restrict__ attn_r, float* __restrict__ el,
    float* __restrict__ er)
{
    const int n = blockIdx.x;
    const int t = threadIdx.x;                 // col 0..127; head = t/32
    float v  = hf[(size_t)n * IN_DIM + t];
    float pl = v * attn_l[t];
    float pr = v * attn_r[t];
#pragma unroll
    for (int off = 16; off > 0; off >>= 1) {
        pl += __shfl_xor(pl, off, 32);
        pr += __shfl_xor(pr, off, 32);
    }
    if ((t & 31) == 0) {
        const int head = t >> 5;
        el[n * HEADS + head] = pl;
        er[n * HEADS + head] = pr;
    }
}

// ---------------------------------------------------------------------------
// K3: init out = 0, m = -FLT_MAX, s = 0 (re-done every call; ws is poisoned)
// ---------------------------------------------------------------------------
__global__ void gat_init(float* __restrict__ out, float* __restrict__ m,
                         float* __restrict__ s, int nout, int nseg)
{
    const int i      = blockIdx.x * blockDim.x + threadIdx.x;
    const int stride = gridDim.x * blockDim.x;
    for (int k = i; k < nout; k += stride) out[k] = 0.0f;
    for (int k = i; k < nseg; k += stride) { m[k] = -3.402823466e38f; s[k] = 0.0f; }
}

// ---------------------------------------------------------------------------
// K4: edge score e = leaky_relu(el[src] + er[dst]); segment-max via atomicMax
// ---------------------------------------------------------------------------
__global__ void gat_edge_score(
    const float* __restrict__ el, const float* __restrict__ er,
    const int* __restrict__ src, const int* __restrict__ dst,
    float* __restrict__ e_out, float* __restrict__ m, long long E4)
{
    const long long i = (long long)blockIdx.x * blockDim.x + threadIdx.x;
    if (i >= E4) return;
    const int edge = (int)(i >> 2);
    const int head = (int)(i & 3);
    const int sN = src[edge], dN = dst[edge];
    float x = el[sN * HEADS + head] + er[dN * HEADS + head];
    x = x > 0.0f ? x : NEG_SLOPE * x;
    e_out[i] = x;
    atomicMax(&m[dN * HEADS + head], x);
}

// ---------------------------------------------------------------------------
// K5: ex = exp(e - m[dst]); segment-sum via atomicAdd (in-place in a-region)
// ---------------------------------------------------------------------------
__global__ void gat_edge_exp(
    const int* __restrict__ dst, float* __restrict__ e_inout,
    const float* __restrict__ m, float* __restrict__ s, long long E4)
{
    const long long i = (long long)blockIdx.x * blockDim.x + threadIdx.x;
    if (i >= E4) return;
    const int edge = (int)(i >> 2);
    const int head = (int)(i & 3);
    const int dN = dst[edge];
    const float ex = expf(e_inout[i] - m[dN * HEADS + head]);
    e_inout[i] = ex;
    atomicAdd(&s[dN * HEADS + head], ex);
}

// ---------------------------------------------------------------------------
// K6: a = ex / s[dst]
// ---------------------------------------------------------------------------
__global__ void gat_edge_norm(
    const int* __restrict__ dst, float* __restrict__ a_inout,
    const float* __restrict__ s, long long E4)
{
    const long long i = (long long)blockIdx.x * blockDim.x + threadIdx.x;
    if (i >= E4) return;
    const int edge = (int)(i >> 2);
    const int head = (int)(i & 3);
    a_inout[i] = a_inout[i] / s[dst[edge] * HEADS + head];
}

// ---------------------------------------------------------------------------
// K7: out[dst] += hf[src] * a.  One wave32 per edge: lane g handles cols
// 4g..4g+3 (float4 gather, 4 f32 atomic adds). hf & out are L2-resident.
// ---------------------------------------------------------------------------
__global__ void __launch_bounds__(256) gat_aggregate(
    const float* __restrict__ hf, const float* __restrict__ a,
    const int* __restrict__ src, const int* __restrict__ dst,
    float* __restrict__ out, long long total)
{
    const long long tid = (long long)blockIdx.x * blockDim.x + threadIdx.x;
    if (tid >= total) return;
    const int edge = (int)(tid >> 5);
    const int g    = (int)(tid & 31);
    const int j    = g * 4;            // column base
    const int head = g >> 3;           // j / 32
    const int sN = src[edge], dN = dst[edge];
    const float av = a[(size_t)edge * HEADS + head];
    const float4 hv = *(const float4*)(hf + (size_t)sN * IN_DIM + j);
    float* op = out + (size_t)dN * IN_DIM + j;
    atomicAdd(op + 0, hv.x * av);
    atomicAdd(op + 1, hv.y * av);
    atomicAdd(op + 2, hv.z * av);
    atomicAdd(op + 3, hv.w * av);
}

// ---------------------------------------------------------------------------
extern "C" void kernel_launch(void* const* d_in, const int* in_sizes, int n_in,
                              void* d_out, int out_size, void* d_ws, size_t ws_size,
                              hipStream_t stream)
{
    const float* h      = (const float*)d_in[0];
    const float* fc_w   = (const float*)d_in[1];
    const float* attn_l = (const float*)d_in[2];
    const float* attn_r = (const float*)d_in[3];
    const int*   src    = (const int*)d_in[4];
    const int*   dst    = (const int*)d_in[5];

    const int N = in_sizes[0] / IN_DIM;   // 50000 (divisible by 16)
    const int E = in_sizes[4];            // 850000

    float* out = (float*)d_out;                        // [N, 128]
    float* a   = (float*)d_out + (size_t)N * IN_DIM;   // [E, 4] (e -> ex -> a)

    float* hf = (float*)d_ws;                          // [N, 128]  25.6 MB
    float* el = hf + (size_t)N * IN_DIM;               // [N, 4]
    float* er = el + (size_t)N * HEADS;                // [N, 4]
    float* m  = er + (size_t)N * HEADS;                // [N, 4]
    float* s  = m  + (size_t)N * HEADS;                // [N, 4]

    gat_gemm_wmma<<<N / 16, 128, 0, stream>>>(h, fc_w, hf);
    gat_elr<<<N, 128, 0, stream>>>(hf, attn_l, attn_r, el, er);
    gat_init<<<1024, 256, 0, stream>>>(out, m, s, N * IN_DIM, N * HEADS);

    const long long E4 = (long long)E * HEADS;
    const int blk = 256;
    const int gE4 = (int)((E4 + blk - 1) / blk);
    gat_edge_score<<<gE4, blk, 0, stream>>>(el, er, src, dst, a, m, E4);
    gat_edge_exp  <<<gE4, blk, 0, stream>>>(dst, a, m, s, E4);
    gat_edge_norm <<<gE4, blk, 0, stream>>>(dst, a, s, E4);

    const long long total = (long long)E * 32;        // one wave per edge
    gat_aggregate<<<(int)((total + blk - 1) / blk), blk, 0, stream>>>(
        hf, a, src, dst, out, total);
}